// ARPrior_3813930959341
// MI455X (gfx1250) — compile-verified
//
#include <hip/hip_runtime.h>
#include <hip/hip_bf16.h>
#include <math.h>

#define B_    64
#define T_    2048
#define ADIM  384
#define NEMB  10
#define HD    3
#define G4    40          // 4*NEMB gate columns
#define NPAD  48          // gate columns padded to 3 WMMA n-tiles
#define MROWS (B_ * T_)   // 131072 rows of the big GEMM

typedef __attribute__((ext_vector_type(2))) float v2f;
typedef __attribute__((ext_vector_type(8))) float v8f;

// ---------------------------------------------------------------------------
// Phase 0a: zero-padded weight copy, Wpad[48][384] (rows 40..47 = 0).
// Removes all predication from the GEMM inner loop.
// ---------------------------------------------------------------------------
__global__ __launch_bounds__(256) void pad_w(const float* __restrict__ W,
                                             float* __restrict__ Wpad) {
    int i = blockIdx.x * 256 + threadIdx.x;     // 0 .. 48*384-1
    if (i >= NPAD * ADIM) return;
    int row = i / ADIM;
    Wpad[i] = (row < G4) ? W[i] : 0.f;
}

// ---------------------------------------------------------------------------
// Phase 0b: fold the qs path into tiny constants.
//   Mq[d][j]  = sum_a qs_W[a][d] * W_ih[j][a]   (3 x 40)
//   qsb[j]    = sum_a qs_b[a]    * W_ih[j][a]   (40)
//   bias[j]   = b_ih[j] + b_hh[j]               (40)
// cbuf layout: [Mq(120) | qsb(40) | bias(40)]
// ---------------------------------------------------------------------------
__global__ void setup_consts(const float* __restrict__ W_ih,
                             const float* __restrict__ b_ih,
                             const float* __restrict__ b_hh,
                             const float* __restrict__ qs_W,
                             const float* __restrict__ qs_b,
                             float* __restrict__ cbuf) {
    int j = threadIdx.x;
    if (j >= G4) return;
    float m0 = 0.f, m1 = 0.f, m2 = 0.f, qb = 0.f;
    for (int a = 0; a < ADIM; ++a) {
        float w = W_ih[j * ADIM + a];
        m0 += qs_W[a * HD + 0] * w;
        m1 += qs_W[a * HD + 1] * w;
        m2 += qs_W[a * HD + 2] * w;
        qb += qs_b[a] * w;
    }
    cbuf[0 * G4 + j] = m0;
    cbuf[1 * G4 + j] = m1;
    cbuf[2 * G4 + j] = m2;
    cbuf[3 * G4 + j] = qb;
    cbuf[4 * G4 + j] = b_ih[j] + b_hh[j];
}

// ---------------------------------------------------------------------------
// Phase 1: pre[m][n] = sum_k X[m][k] * Wpad[n][k]  via V_WMMA_F32_16X16X4_F32.
// One wave owns one 16-row strip and all 3 n-tiles; inner loop is
// 4 unconditional b64 loads + 3 WMMAs, no exec manipulation.
// A (16x4 f32): lane<16 -> row=lane, K=0..1 ; lane>=16 -> row=lane-16, K=2..3.
// B (4x16 f32): mirrored half-wave K split, N = lane&15 (+ 16*tile).
// C (16x16 f32): VGPR r -> row r + 8*(lane>=16), col lane&15.
// ---------------------------------------------------------------------------
__global__ __launch_bounds__(256) void gemm_pre(const float* __restrict__ X,
                                                const float* __restrict__ Wpad,
                                                float* __restrict__ pre) {
    const int wave = threadIdx.x >> 5;
    const int lane = threadIdx.x & 31;
    const int mt   = blockIdx.x * 8 + wave;   // 0..8191
    const int m0   = mt * 16;
    const int half = lane >> 4;               // K sub-offset selector
    const int lp   = lane & 15;

    const float* arow = X    + (size_t)(m0 + lp) * ADIM + 2 * half;
    const float* brow = Wpad + (size_t)lp        * ADIM + 2 * half;

    v8f c0 = {}, c1 = {}, c2 = {};
    for (int k = 0; k < ADIM; k += 4) {
        v2f a  = *(const v2f*)(arow + k);
        v2f b0 = *(const v2f*)(brow + k);
        v2f b1 = *(const v2f*)(brow + 16 * ADIM + k);
        v2f b2 = *(const v2f*)(brow + 32 * ADIM + k);
        c0 = __builtin_amdgcn_wmma_f32_16x16x4_f32(false, a, false, b0,
                                                   (short)0, c0, false, false);
        c1 = __builtin_amdgcn_wmma_f32_16x16x4_f32(false, a, false, b1,
                                                   (short)0, c1, false, false);
        c2 = __builtin_amdgcn_wmma_f32_16x16x4_f32(false, a, false, b2,
                                                   (short)0, c2, false, false);
    }

#pragma unroll
    for (int r = 0; r < 8; ++r) {
        size_t row = (size_t)(m0 + r + 8 * half);
        float* p = pre + row * NPAD + lp;
        p[0]  = c0[r];
        p[16] = c1[r];
        p[32] = c2[r];
    }
}

// ---------------------------------------------------------------------------
// Phase 2: per-batch-row recurrence. One block (64 threads, 2 waves) per row.
// Per step: gates from precomputed pre + tiny LDS-resident matvecs,
// LSTM activations, log-softmax/argmax over 10 logits, codebook gather.
// ---------------------------------------------------------------------------
__global__ __launch_bounds__(64) void recur(const float* __restrict__ pre,
                                            const int*   __restrict__ inds,
                                            const float* __restrict__ codebook,
                                            const float* __restrict__ W_hh,
                                            const float* __restrict__ cbuf,
                                            float* __restrict__ out_emb,
                                            float* __restrict__ partials) {
    const int b = blockIdx.x;
    const int j = threadIdx.x;   // 0..63

    __shared__ float sWhh[G4 * NEMB];
    __shared__ float sMq[3 * G4], sQsb[G4], sBias[G4], sCb[NEMB * HD];
    __shared__ float sH[NEMB], sEmb[HD], sAct[G4];

    for (int i = j; i < G4 * NEMB; i += 64) sWhh[i] = W_hh[i];
    for (int i = j; i < 3 * G4; i += 64) sMq[i] = cbuf[i];
    if (j < G4) { sQsb[j] = cbuf[3 * G4 + j]; sBias[j] = cbuf[4 * G4 + j]; }
    if (j < NEMB * HD) sCb[j] = codebook[j];
    if (j < NEMB) sH[j] = 0.f;
    if (j < HD)   sEmb[j] = 0.f;
    __syncthreads();

    float cc = 0.f;        // cell state of unit j (threads 0..9)
    float lossAcc = 0.f;   // thread 0: sum of picked log-probs

    const float* prow = pre     + (size_t)b * T_ * NPAD;
    const int*   irow = inds    + (size_t)b * T_;
    float*       orow = out_emb + (size_t)b * T_ * HD;

    for (int t = 0; t < T_; ++t) {
        if (t + 1 < T_)
            __builtin_prefetch(prow + (size_t)(t + 1) * NPAD, 0, 0);

        if (j < G4) {
            float g = prow[(size_t)t * NPAD + j] + sBias[j];
            if (t > 0)
                g += sQsb[j] + sEmb[0] * sMq[j] + sEmb[1] * sMq[G4 + j]
                             + sEmb[2] * sMq[2 * G4 + j];
            float acc = 0.f;
#pragma unroll
            for (int k = 0; k < NEMB; ++k) acc += sH[k] * sWhh[j * NEMB + k];
            g += acc;
            float a;
            if (j < 2 * NEMB || j >= 3 * NEMB)          // i, f, o gates
                a = 1.f / (1.f + __expf(-g));
            else                                         // g gate
                a = tanhf(g);
            sAct[j] = a;
        }
        __syncthreads();

        if (j < NEMB) {
            cc = sAct[NEMB + j] * cc + sAct[j] * sAct[2 * NEMB + j];
            sH[j] = sAct[3 * NEMB + j] * tanhf(cc);
        }
        __syncthreads();

        if (j < 4) {  // redundant tiny softmax/argmax on a few lanes
            float m = sH[0];
            int am = 0;
#pragma unroll
            for (int k = 1; k < NEMB; ++k) {
                float v = sH[k];
                if (v > m) { m = v; am = k; }           // first-max, like argmax
            }
            float s = 0.f;
#pragma unroll
            for (int k = 0; k < NEMB; ++k) s += __expf(sH[k] - m);
            float lse = m + __logf(s);
            if (j == 0) {
                int gt = irow[t];
                lossAcc += sH[gt] - lse;                // picked log-prob
            }
            if (j < HD) {
                float e = sCb[am * HD + j];
                sEmb[j] = e;
                orow[(size_t)t * HD + j] = e;
            }
        }
        __syncthreads();
    }
    if (j == 0) partials[b] = lossAcc;
}

// ---------------------------------------------------------------------------
// Phase 3: deterministic loss reduction (overwrites, no atomics on d_out).
// ---------------------------------------------------------------------------
__global__ void loss_reduce(const float* __restrict__ partials,
                            float* __restrict__ out_loss) {
    if (threadIdx.x == 0) {
        float s = 0.f;
        for (int i = 0; i < B_; ++i) s += partials[i];
        *out_loss = -s / (float)(B_ * T_);
    }
}

extern "C" void kernel_launch(void* const* d_in, const int* in_sizes, int n_in,
                              void* d_out, int out_size, void* d_ws, size_t ws_size,
                              hipStream_t stream) {
    const float* X    = (const float*)d_in[0];
    const int*   inds = (const int*)d_in[1];
    const float* cb   = (const float*)d_in[2];
    const float* W_ih = (const float*)d_in[3];
    const float* W_hh = (const float*)d_in[4];
    const float* b_ih = (const float*)d_in[5];
    const float* b_hh = (const float*)d_in[6];
    const float* qs_W = (const float*)d_in[7];
    const float* qs_b = (const float*)d_in[8];

    float* pre      = (float*)d_ws;                       // 131072*48 f32 (~25 MB)
    float* Wpad     = pre + (size_t)MROWS * NPAD;         // 48*384 f32 (74 KB)
    float* cbuf     = Wpad + NPAD * ADIM;                 // 5*40 f32
    float* partials = cbuf + 5 * G4;                      // 64 f32

    float* out_emb  = (float*)d_out;                      // (B,T,3)
    float* out_loss = out_emb + (size_t)B_ * T_ * HD;     // scalar

    pad_w<<<(NPAD * ADIM + 255) / 256, 256, 0, stream>>>(W_ih, Wpad);
    setup_consts<<<1, 64, 0, stream>>>(W_ih, b_ih, b_hh, qs_W, qs_b, cbuf);
    gemm_pre<<<MROWS / 16 / 8, 256, 0, stream>>>(X, Wpad, pre);
    recur<<<B_, 64, 0, stream>>>(pre, inds, cb, W_hh, cbuf, out_emb, partials);
    loss_reduce<<<1, 32, 0, stream>>>(partials, out_loss);
}